// SGC_16587163697543
// MI455X (gfx1250) — compile-verified
//
#include <hip/hip_runtime.h>
#include <hip/hip_bf16.h>

// SGC (4-layer SGConv GNN) for MI455X / gfx1250, wave32.
//
// Strategy (see analysis):
//  - norm[n]   = rsqrt(in_degree incl. self-loop)         (float atomics)
//  - h         = row-L2-normalized x                      (wave32 shfl reduce)
//  - per layer: h *= norm (in place), agg = h (self-loop),
//               agg[dst] += h[src]  (global_atomic_add_f32, L2-resident),
//               out = relu((agg .* norm) @ W^T + b)  via V_WMMA_F32_16X16X4_F32.
//
// Working set (h, agg: 51 MB each) fits in the 192 MB L2 -> scatter is
// L2-bound. GEMM FLOPs are trivial; f32 WMMA keeps reference precision.

typedef __attribute__((ext_vector_type(2))) float v2f;
typedef __attribute__((ext_vector_type(8))) float v8f;

#define IN_F 128
#define HID_F 128

// ---------------- degree / norm ----------------

__global__ __launch_bounds__(256) void sgc_init_deg(float* __restrict__ norm, int N) {
  int i = blockIdx.x * 256 + threadIdx.x;
  if (i < N) norm[i] = 1.0f;  // self-loop contributes 1 to in-degree
}

__global__ __launch_bounds__(256) void sgc_count_deg(const int* __restrict__ dst,
                                                     float* __restrict__ norm, int E) {
  int e = blockIdx.x * 256 + threadIdx.x;
  if (e < E) atomicAdd(&norm[dst[e]], 1.0f);
}

__global__ __launch_bounds__(256) void sgc_finalize_norm(float* __restrict__ norm, int N) {
  int i = blockIdx.x * 256 + threadIdx.x;
  if (i < N) norm[i] = rsqrtf(fmaxf(norm[i], 1.0f));
}

// ---------------- row-wise L2 normalize: h = x / max(||x||, eps) -------------
// One wave per row; each lane owns a float4 (32 lanes * 4 = 128 features).

__global__ __launch_bounds__(256) void sgc_l2norm_rows(const float* __restrict__ x,
                                                       float* __restrict__ h, int N) {
  int row = blockIdx.x * 8 + threadIdx.y;
  if (row >= N) return;
  int lane = threadIdx.x;
  float4 v = reinterpret_cast<const float4*>(x + (size_t)row * IN_F)[lane];
  float ss = v.x * v.x + v.y * v.y + v.z * v.z + v.w * v.w;
#pragma unroll
  for (int off = 16; off > 0; off >>= 1) ss += __shfl_xor(ss, off, 32);
  float inv = 1.0f / fmaxf(sqrtf(ss), 1e-12f);
  v.x *= inv; v.y *= inv; v.z *= inv; v.w *= inv;
  reinterpret_cast<float4*>(h + (size_t)row * IN_F)[lane] = v;
}

// ---------------- h *= norm (in place) ; agg = h (self-loop init) ------------

__global__ __launch_bounds__(256) void sgc_scale_selfloop(float* __restrict__ h,
                                                          float* __restrict__ agg,
                                                          const float* __restrict__ norm,
                                                          int N) {
  int idx = blockIdx.x * 256 + threadIdx.x;  // float4 index
  int total = N * (HID_F / 4);
  if (idx >= total) return;
  int row = idx >> 5;  // 32 float4 per row
  float s = norm[row];
  float4 v = reinterpret_cast<float4*>(h)[idx];
  v.x *= s; v.y *= s; v.z *= s; v.w *= s;
  reinterpret_cast<float4*>(h)[idx] = v;
  reinterpret_cast<float4*>(agg)[idx] = v;
}

// ---------------- edge scatter: agg[dst,:] += h[src,:] -----------------------
// One wave per edge; float4 gather per lane + 4 non-returning f32 atomics.

__global__ __launch_bounds__(256) void sgc_scatter_edges(const int* __restrict__ src,
                                                         const int* __restrict__ dst,
                                                         const float* __restrict__ h,
                                                         float* __restrict__ agg, int E) {
  int e = blockIdx.x * 8 + threadIdx.y;
  if (e >= E) return;
  int lane = threadIdx.x;
  int s = src[e];
  int d = dst[e];
  float4 v = reinterpret_cast<const float4*>(h + (size_t)s * HID_F)[lane];
  float* dp = agg + (size_t)d * HID_F + lane * 4;
  atomicAdd(dp + 0, v.x);
  atomicAdd(dp + 1, v.y);
  atomicAdd(dp + 2, v.z);
  atomicAdd(dp + 3, v.w);
}

// ---------------- fused epilogue GEMM: out = act((agg .* norm) @ W^T + b) ----
// One wave computes one 16x16 C tile, K = 128 via 32 x V_WMMA_F32_16X16X4_F32.
// A 16x4 f32 layout (ISA 7.12.2): lane%16 = M row; lanes 0-15 hold K{0,1},
// lanes 16-31 hold K{2,3} -> contiguous float2 per lane. B (= W^T tile)
// mirrors with lane%16 = N column, contiguous in K from row-major W.
// C/D: VGPR r -> (M = r + 8*(lane/16), N = lane%16). Bias seeds C; the
// row-scale by norm is fused into the A load. Tile index is wave-uniform, so
// EXEC is all ones at every WMMA (required by ISA).

template <int FOUT, bool RELU>
__global__ __launch_bounds__(256) void sgc_gemm_wmma(const float* __restrict__ agg,
                                                     const float* __restrict__ norm,
                                                     const float* __restrict__ W,
                                                     const float* __restrict__ bias,
                                                     float* __restrict__ out, int nRows) {
  const int lane = threadIdx.x;        // 0..31
  const int ntiles = FOUT / 16;
  const int tile = blockIdx.x * 8 + threadIdx.y;   // wave-uniform
  const int mtiles = nRows / 16;
  if (tile >= mtiles * ntiles) return;
  const int mt = tile / ntiles;
  const int nt = tile % ntiles;
  const int half = lane >> 4;          // 0: K+{0,1}, 1: K+{2,3}
  const int l16 = lane & 15;
  const int arow = mt * 16 + l16;      // A row handled by this lane
  const int col = nt * 16 + l16;       // B/C column handled by this lane

  const float nrm = norm[arow];
  const float bv = bias[col];
  v8f c;
#pragma unroll
  for (int i = 0; i < 8; ++i) c[i] = bv;   // D = A*B + C with C = bias

  const float* __restrict__ ap = agg + (size_t)arow * HID_F;
  const float* __restrict__ wp = W + (size_t)col * HID_F;  // W is [FOUT][128]

#pragma unroll
  for (int k0 = 0; k0 < HID_F; k0 += 4) {
    const int k = k0 + 2 * half;
    v2f a, b;
    a.x = ap[k] * nrm;
    a.y = ap[k + 1] * nrm;
    b.x = wp[k];
    b.y = wp[k + 1];
    c = __builtin_amdgcn_wmma_f32_16x16x4_f32(false, a, false, b, (short)0, c,
                                              false, false);
  }

#pragma unroll
  for (int r = 0; r < 8; ++r) {
    int m = mt * 16 + r + 8 * half;
    float v = c[r];
    if (RELU) v = fmaxf(v, 0.0f);
    out[(size_t)m * FOUT + col] = v;
  }
}

// ---------------- host-side orchestration -----------------------------------

extern "C" void kernel_launch(void* const* d_in, const int* in_sizes, int n_in,
                              void* d_out, int out_size, void* d_ws, size_t ws_size,
                              hipStream_t stream) {
  const float* x = (const float*)d_in[0];
  const int* src = (const int*)d_in[1];
  const int* dst = (const int*)d_in[2];
  const float* Wm[4] = {(const float*)d_in[3], (const float*)d_in[5],
                        (const float*)d_in[7], (const float*)d_in[9]};
  const float* Bv[4] = {(const float*)d_in[4], (const float*)d_in[6],
                        (const float*)d_in[8], (const float*)d_in[10]};

  const int N = in_sizes[0] / IN_F;  // 100000
  const int E = in_sizes[1];         // 1600000

  // workspace layout: norm | h | agg   (~103 MB total)
  char* ws = (char*)d_ws;
  float* norm = (float*)ws;
  float* h = (float*)(ws + 512 * 1024);
  float* agg = h + (size_t)N * HID_F;

  dim3 wave8(32, 8);

  // degrees -> norm
  sgc_init_deg<<<(N + 255) / 256, 256, 0, stream>>>(norm, N);
  sgc_count_deg<<<(E + 255) / 256, 256, 0, stream>>>(dst, norm, E);
  sgc_finalize_norm<<<(N + 255) / 256, 256, 0, stream>>>(norm, N);

  // h = L2-normalized x
  sgc_l2norm_rows<<<(N + 7) / 8, wave8, 0, stream>>>(x, h, N);

  const int n4 = N * (HID_F / 4);
  for (int layer = 0; layer < 4; ++layer) {
    sgc_scale_selfloop<<<(n4 + 255) / 256, 256, 0, stream>>>(h, agg, norm, N);
    sgc_scatter_edges<<<(E + 7) / 8, wave8, 0, stream>>>(src, dst, h, agg, E);
    if (layer < 3) {
      int tiles = (N / 16) * (HID_F / 16);
      sgc_gemm_wmma<HID_F, true><<<(tiles + 7) / 8, wave8, 0, stream>>>(
          agg, norm, Wm[layer], Bv[layer], h, N);
    } else {
      int tiles = (N / 16) * (64 / 16);
      sgc_gemm_wmma<64, false><<<(tiles + 7) / 8, wave8, 0, stream>>>(
          agg, norm, Wm[3], Bv[3], (float*)d_out, N);
    }
  }
}